// ConfidenceWeightedMLLWithTies_21380347199790
// MI455X (gfx1250) — compile-verified
//
#include <hip/hip_runtime.h>
#include <hip/hip_bf16.h>
#include <math.h>

typedef __attribute__((ext_vector_type(2))) float v2f;
typedef __attribute__((ext_vector_type(8))) float v8f;

#define NBLOCKS 1024
#define NTHREADS 256

// Full-wave (32 lane) sum via V_WMMA_F32_16X16X4_F32.
// A 16x4 f32 layout: lane L (0..15) -> row M=L, VGPR0=A[L][0], VGPR1=A[L][1];
// lane L+16 -> same row M=L, VGPR0=A[L][2], VGPR1=A[L][3].
// With a0 = v(lane), a1 = 0: row L sum = v_L + v_{L+16}.
// B = all ones (4x16, layout-independent), C = 0 => D[m][n] = rowsum_m.
// D layout: lane L (0..15) VGPR r = D[r][L]; lane L+16 VGPR r = D[8+r][L].
// Lane-local sum of 8 D regs + one xor-16 shuffle -> full wave total in all lanes.
// Requires EXEC == all ones (call only from uniform control flow, full waves).
__device__ __forceinline__ float wave_sum_wmma(float v) {
    v2f a; a.x = v;    a.y = 0.0f;
    v2f b; b.x = 1.0f; b.y = 1.0f;
    v8f c = {};
    v8f d = __builtin_amdgcn_wmma_f32_16x16x4_f32(
        /*neg_a=*/false, a, /*neg_b=*/false, b,
        /*c_mod=*/(short)0, c, /*reuse_a=*/false, /*reuse_b=*/false);
    float s = ((d[0] + d[1]) + (d[2] + d[3])) + ((d[4] + d[5]) + (d[6] + d[7]));
    s += __shfl_xor(s, 16, 32);
    return s;
}

__device__ __forceinline__ float phi_from_erf_arg(float z) {
    // phi(z) = 0.5*(1+erf(z/sqrt(2)))
    return fmaf(0.5f, erff(z * 0.70710678118654752440f), 0.5f);
}

// One comparison: gathers + 2 erf + 1 log, accumulate T += w*lp, S += w.
__device__ __forceinline__ void process_one(const float* __restrict__ mean,
                                            const float* __restrict__ var,
                                            int a, int b, int c, float wi,
                                            float& t_acc, float& s_acc) {
    const float tol = 0.1f;
    const float md  = mean[a] - mean[b];
    const float vd  = var[a] + var[b];
    const float inv = rsqrtf(vd + 1e-6f);   // 1/std

    const float ad  = fabsf(md);
    // erf slot A: type0 -> (md - tol), type1 -> (-md - tol), type2 -> (tol - ad)
    const float sgn = (c == 1) ? -md : md;
    const float za  = (c == 2) ? (tol - ad) : (sgn - tol);
    // erf slot B: only meaningful for ties: (-tol - ad)
    const float zb  = -tol - ad;

    const float pa = phi_from_erf_arg(za * inv);
    const float pb = phi_from_erf_arg(zb * inv);
    const float p  = (c == 2) ? (pa - pb) : pa;
    const float lp = logf(p + 1e-8f);

    t_acc += wi * lp;
    s_acc += wi;
}

// Pass 1: fused gather + log-prob + dual reduction (T = sum(w*lp), S = sum(w)).
// Streamed arrays consumed 4 elements/thread via b128 loads; 4 independent
// erf/log chains per iteration hide TRANS-pipe latency.
__global__ void __launch_bounds__(NTHREADS)
cwmll_partial(const float* __restrict__ mean,
              const float* __restrict__ var,
              const float* __restrict__ w,
              const int*   __restrict__ ia,
              const int*   __restrict__ ib,
              const int*   __restrict__ ct,
              float2* __restrict__ partial,
              int m_total) {
    float t_acc = 0.0f, s_acc = 0.0f;
    const int gtid     = blockIdx.x * blockDim.x + threadIdx.x;
    const int nthreads = gridDim.x * blockDim.x;
    const int m4       = m_total >> 2;

    const int4*   ia4 = (const int4*)ia;
    const int4*   ib4 = (const int4*)ib;
    const int4*   ct4 = (const int4*)ct;
    const float4* w4p = (const float4*)w;

    for (int i = gtid; i < m4; i += nthreads) {
        const int4   a4 = ia4[i];
        const int4   b4 = ib4[i];
        const int4   c4 = ct4[i];
        const float4 w4 = w4p[i];
        process_one(mean, var, a4.x, b4.x, c4.x, w4.x, t_acc, s_acc);
        process_one(mean, var, a4.y, b4.y, c4.y, w4.y, t_acc, s_acc);
        process_one(mean, var, a4.z, b4.z, c4.z, w4.z, t_acc, s_acc);
        process_one(mean, var, a4.w, b4.w, c4.w, w4.w, t_acc, s_acc);
    }
    // Tail (m_total % 4 elements), <= 3 threads active.
    for (int i = (m4 << 2) + gtid; i < m_total; i += nthreads) {
        process_one(mean, var, ia[i], ib[i], ct[i], w[i], t_acc, s_acc);
    }

    // Uniform control flow from here; full waves -> EXEC all ones for WMMA.
    t_acc = wave_sum_wmma(t_acc);
    s_acc = wave_sum_wmma(s_acc);

    __shared__ float2 lds[NTHREADS / 32];
    const int wid  = threadIdx.x >> 5;
    const int lane = threadIdx.x & 31;
    if (lane == 0) lds[wid] = make_float2(t_acc, s_acc);
    __syncthreads();
    if (threadIdx.x == 0) {
        float tt = 0.0f, ss = 0.0f;
        #pragma unroll
        for (int k = 0; k < NTHREADS / 32; ++k) { tt += lds[k].x; ss += lds[k].y; }
        partial[blockIdx.x] = make_float2(tt, ss);
    }
}

// Pass 2: reduce block partials, apply scalar normalization (M/s).
__global__ void __launch_bounds__(NTHREADS)
cwmll_final(const float2* __restrict__ partial, int nblocks,
            float* __restrict__ out, float m_as_float) {
    float t_acc = 0.0f, s_acc = 0.0f;
    for (int i = threadIdx.x; i < nblocks; i += blockDim.x) {
        const float2 p = partial[i];
        t_acc += p.x;
        s_acc += p.y;
    }
    t_acc = wave_sum_wmma(t_acc);
    s_acc = wave_sum_wmma(s_acc);

    __shared__ float2 lds[NTHREADS / 32];
    const int wid  = threadIdx.x >> 5;
    const int lane = threadIdx.x & 31;
    if (lane == 0) lds[wid] = make_float2(t_acc, s_acc);
    __syncthreads();
    if (threadIdx.x == 0) {
        float tt = 0.0f, ss = 0.0f;
        #pragma unroll
        for (int k = 0; k < NTHREADS / 32; ++k) { tt += lds[k].x; ss += lds[k].y; }
        out[0] = (ss > 0.0f) ? (tt * m_as_float / ss) : tt;
    }
}

extern "C" void kernel_launch(void* const* d_in, const int* in_sizes, int n_in,
                              void* d_out, int out_size, void* d_ws, size_t ws_size,
                              hipStream_t stream) {
    const float* mean = (const float*)d_in[0];
    const float* var  = (const float*)d_in[1];
    const float* w    = (const float*)d_in[2];
    const int*   ia   = (const int*)d_in[3];
    const int*   ib   = (const int*)d_in[4];
    const int*   ct   = (const int*)d_in[5];
    float* out = (float*)d_out;

    const int m_total = in_sizes[2];
    float2* partial = (float2*)d_ws;   // NBLOCKS * sizeof(float2) = 8 KB

    cwmll_partial<<<NBLOCKS, NTHREADS, 0, stream>>>(mean, var, w, ia, ib, ct,
                                                    partial, m_total);
    cwmll_final<<<1, NTHREADS, 0, stream>>>(partial, NBLOCKS, out, (float)m_total);
}